// CTPN_Loss_30073361006848
// MI455X (gfx1250) — compile-verified
//
#include <hip/hip_runtime.h>
#include <hip/hip_bf16.h>

// CTPN loss on MI455X (gfx1250, wave32).
// Latency-bound gather + deterministic reduction; WMMA f32 16x16x4 used as the
// in-wave cross-lane summation engine (B = ones => D[m,n] = rowsum_m of A).

#define H_DIM 512
#define W_DIM 1024
#define K_DIM 10
#define HW (H_DIM * W_DIM)

typedef __attribute__((ext_vector_type(2))) float v2f;
typedef __attribute__((ext_vector_type(8))) float v8f;

// Sum one float across all 32 lanes of the wave, deterministically.
// Layout reasoning (documented in CDNA5 ISA):
//   A 16x4 f32: lane m (m<16) holds A[m,0..1] in vgpr0..1; lane m+16 holds A[m,2..3].
//   We place partial in vgpr0, 0 in vgpr1 -> rowsum_m = p(lane m) + p(lane m+16).
//   D 16x16: vgpr v, lanes 0-15 = row v, lanes 16-31 = row v+8; D[m,n]=rowsum_m (all n).
//   Per-lane sum of D vgpr0..7: lanes 0-15 = sum rows 0..7, lanes 16-31 = sum rows 8..15.
//   One xor-16 shuffle + add yields the full 32-lane total in every lane.
__device__ __forceinline__ float wave32_sum_wmma(float p) {
    v2f a; a[0] = p;    a[1] = 0.0f;
    v2f b; b[0] = 1.0f; b[1] = 1.0f;   // all-ones B: layout-agnostic
    v8f c = {};
    v8f d = __builtin_amdgcn_wmma_f32_16x16x4_f32(
        /*neg_a=*/false, a, /*neg_b=*/false, b,
        /*c_mod=*/(short)0, c, /*reuse_a=*/false, /*reuse_b=*/false);
    float s = ((d[0] + d[1]) + (d[2] + d[3])) + ((d[4] + d[5]) + (d[6] + d[7]));
    s += __shfl_xor(s, 16, 32);
    return s;
}

__device__ __forceinline__ float smooth_l1(float d) {
    float ad = fabsf(d);
    return (ad < 1.0f) ? 0.5f * d * d : ad - 0.5f;
}

__global__ void __launch_bounds__(256, 1)
ctpn_loss_kernel(const float* __restrict__ score,
                 const float* __restrict__ vert,
                 const float* __restrict__ side,
                 const int*   __restrict__ pos_yx,
                 const int*   __restrict__ pos_k,
                 const int*   __restrict__ neg_yx,
                 const int*   __restrict__ neg_k,
                 const int*   __restrict__ v_yx,
                 const int*   __restrict__ v_k,
                 const float* __restrict__ v_target,
                 const int*   __restrict__ o_yx,
                 const int*   __restrict__ o_k,
                 const float* __restrict__ o_target,
                 float*       __restrict__ out)
{
    __shared__ float wsum[8][3];
    const int tid = threadIdx.x;

    float cls_p = 0.0f, v_p = 0.0f, o_p = 0.0f;

    // ---- classification: 64 pos + 64 neg, one item per thread (tid < 128) ----
    if (tid < 128) {
        const bool is_pos = (tid < 64);
        const int  i      = is_pos ? tid : tid - 64;
        const int* yx     = is_pos ? pos_yx : neg_yx;
        const int* kk     = is_pos ? pos_k  : neg_k;
        const int y = yx[2 * i];
        const int x = yx[2 * i + 1];
        const int k = kk[i];
        const int base = (2 * k) * HW + y * W_DIM + x;     // channel (k,0)
        const float l0 = score[base];
        const float l1 = score[base + HW];                  // channel (k,1)
        const float m   = fmaxf(l0, l1);
        const float lse = m + logf(expf(l0 - m) + expf(l1 - m));
        cls_p = lse - (is_pos ? l1 : l0);                   // -log_softmax[cls]
    }

    // ---- vertical regression: 512 samples x 2 comps, 2 samples per thread ----
    #pragma unroll
    for (int s = tid; s < 512; s += 256) {
        const int y = v_yx[2 * s];
        const int x = v_yx[2 * s + 1];
        const int k = v_k[s];
        const int base = (2 * k) * HW + y * W_DIM + x;
        v_p += smooth_l1(vert[base]      - v_target[2 * s]);
        v_p += smooth_l1(vert[base + HW] - v_target[2 * s + 1]);
    }

    // ---- side refinement: 256 samples, one per thread ----
    {
        const int y = o_yx[2 * tid];
        const int x = o_yx[2 * tid + 1];
        const int k = o_k[tid];
        o_p = smooth_l1(side[k * HW + y * W_DIM + x] - o_target[tid]);
    }

    // ---- deterministic reduction: WMMA wave sums, then 8 wave partials ----
    const float cls_w = wave32_sum_wmma(cls_p);
    const float v_w   = wave32_sum_wmma(v_p);
    const float o_w   = wave32_sum_wmma(o_p);

    const int wave = tid >> 5;
    const int lane = tid & 31;
    if (lane == 0) {
        wsum[wave][0] = cls_w;
        wsum[wave][1] = v_w;
        wsum[wave][2] = o_w;
    }
    __syncthreads();

    if (tid == 0) {
        float c = 0.0f, vv = 0.0f, oo = 0.0f;
        #pragma unroll
        for (int w = 0; w < 8; ++w) {
            c  += wsum[w][0];
            vv += wsum[w][1];
            oo += wsum[w][2];
        }
        c  *= (1.0f / 128.0f);    // / NS
        vv *= (1.0f / 1024.0f);   // mean over (512,2)
        oo *= (1.0f / 256.0f);    // mean over (256,)
        out[0] = c + vv + oo;     // loss (lambda1 = lambda2 = 1)
        out[1] = c;               // cls_loss
        out[2] = vv;              // v_reg_loss
        out[3] = oo;              // o_reg_loss
    }
}

extern "C" void kernel_launch(void* const* d_in, const int* in_sizes, int n_in,
                              void* d_out, int out_size, void* d_ws, size_t ws_size,
                              hipStream_t stream) {
    (void)in_sizes; (void)n_in; (void)out_size; (void)d_ws; (void)ws_size;
    ctpn_loss_kernel<<<1, 256, 0, stream>>>(
        (const float*)d_in[0],   // score (1, 2K, H, W)
        (const float*)d_in[1],   // vertical_pred (1, 2K, H, W)
        (const float*)d_in[2],   // side_refinement (1, K, H, W)
        (const int*)d_in[3],     // pos_yx (64,2)
        (const int*)d_in[4],     // pos_k (64,)
        (const int*)d_in[5],     // neg_yx (64,2)
        (const int*)d_in[6],     // neg_k (64,)
        (const int*)d_in[7],     // v_yx (512,2)
        (const int*)d_in[8],     // v_k (512,)
        (const float*)d_in[9],   // v_target (512,2)
        (const int*)d_in[10],    // o_yx (256,2)
        (const int*)d_in[11],    // o_k (256,)
        (const float*)d_in[12],  // o_target (256,)
        (float*)d_out);
}